// DifferentiableReRanker_37864431682215
// MI455X (gfx1250) — compile-verified
//
#include <hip/hip_runtime.h>

typedef float v2f __attribute__((ext_vector_type(2)));
typedef float v8f __attribute__((ext_vector_type(8)));

#define NN  256
#define BB  512
#define EPS 0.1f

// ---------------------------------------------------------------------------
// Kernel 1: refinement MLP via V_WMMA_F32_16X16X4_F32.
// refined[m] = sum_k relu(s_m*W1[k]+b1[k]) * W2[k] + b2
// One block of 512 threads = 16 waves; wave w computes rows m = 16w..16w+15.
// A (16x4 f32 layout): lane L<16 -> {A[m][0],A[m][1]}, lane>=16 -> {A[m][2],A[m][3]}
// B (4x16): all columns identical (W2 broadcast), so only the K mapping matters.
// C/D (16x16 f32): VGPR r, lanes 0-15 -> M=r, lanes 16-31 -> M=8+r; any column N works.
// ---------------------------------------------------------------------------
__global__ __launch_bounds__(512) void mlp_refine_wmma(
    const float* __restrict__ cross, const float* __restrict__ W1,
    const float* __restrict__ b1,    const float* __restrict__ W2,
    const float* __restrict__ b2,    float* __restrict__ refined)
{
    const int tid  = threadIdx.x;
    const int wave = tid >> 5;            // 0..15 : row tile
    const int lane = tid & 31;
    const int mloc = lane & 15;
    const int kb   = (lane < 16) ? 0 : 2; // K base for this half-wave
    const int m    = wave * 16 + mloc;

    const float s = cross[m];

    v8f c = {0.f, 0.f, 0.f, 0.f, 0.f, 0.f, 0.f, 0.f};
#pragma unroll
    for (int kc = 0; kc < 8; ++kc) {      // K = 32 as 8 chunks of 4
        const int k0 = kc * 4 + kb;
        float h0 = fmaf(s, W1[k0],     b1[k0]);
        float h1 = fmaf(s, W1[k0 + 1], b1[k0 + 1]);
        v2f a, bm;
        a.x  = h0 > 0.f ? h0 : 0.f;
        a.y  = h1 > 0.f ? h1 : 0.f;
        bm.x = W2[k0];
        bm.y = W2[k0 + 1];
        c = __builtin_amdgcn_wmma_f32_16x16x4_f32(
                /*neg_a=*/false, a, /*neg_b=*/false, bm,
                /*c_mod=*/(short)0, c, /*reuse_a=*/false, /*reuse_b=*/false);
    }

    const float bias = b2[0];
    if (lane == 0) {            // N=0 column, M = 0..7
#pragma unroll
        for (int r = 0; r < 8; ++r) refined[wave * 16 + r] = c[r] + bias;
    } else if (lane == 16) {    // N=0 column, M = 8..15
#pragma unroll
        for (int r = 0; r < 8; ++r) refined[wave * 16 + 8 + r] = c[r] + bias;
    }
}

// ---------------------------------------------------------------------------
// Kernel 2: per-row descending bitonic sort + two PAVA isotonic regressions.
// One block (256 threads, 8 wave32s) per row. PAVA A runs on thread 0 (wave 0),
// PAVA B on thread 32 (wave 1) so the two serial passes overlap.
// soft_sort:  out1[i] = w[i] - iso_noninc(w - s)[i],      w[i] = (N-i)/eps
// soft_rank:  z_desc[i] = -s[N-1-i]/eps, sigma_z[i] = idx[N-1-i]
//             out2[sigma_z[i]] = z_desc[i] - iso_noninc(z_desc - rho)[i]
// ---------------------------------------------------------------------------
__global__ __launch_bounds__(256) void sort_pava_kernel(
    const float* __restrict__ refined, const float* __restrict__ init,
    float* __restrict__ out)
{
    __shared__ float skey[NN];
    __shared__ int   sidx[NN];
    __shared__ float v1[NN];
    __shared__ float v2[NN];
    __shared__ float psumA[NN];  __shared__ int pcntA[NN];
    __shared__ float psumB[NN];  __shared__ int pcntB[NN];

    const int b = blockIdx.x;
    const int t = threadIdx.x;

    skey[t] = refined[t] + 0.1f * init[b * NN + t];
    sidx[t] = t;
    __syncthreads();

    // ---- bitonic sort, descending, with index payload ----
    for (int k = 2; k <= NN; k <<= 1) {
        for (int j = k >> 1; j > 0; j >>= 1) {
            int ixj = t ^ j;
            if (ixj > t) {
                float a = skey[t], c = skey[ixj];
                bool dn = ((t & k) == 0);                 // descending segment
                bool sw = dn ? (a < c) : (a > c);
                if (sw) {
                    skey[t] = c; skey[ixj] = a;
                    int ia = sidx[t]; sidx[t] = sidx[ixj]; sidx[ixj] = ia;
                }
            }
            __syncthreads();
        }
    }

    // ---- PAVA (nonincreasing fit): merge while mean(top) > mean(prev) ----
    if (t == 0) {
        int top = -1;
        for (int i = 0; i < NN; ++i) {
            float yi = (float)(NN - i) / EPS - skey[i];
            ++top; psumA[top] = yi; pcntA[top] = 1;
            while (top > 0 &&
                   psumA[top] * (float)pcntA[top - 1] >
                   psumA[top - 1] * (float)pcntA[top]) {
                psumA[top - 1] += psumA[top];
                pcntA[top - 1] += pcntA[top];
                --top;
            }
        }
        int pos = 0;
        for (int blk = 0; blk <= top; ++blk) {
            float mv = psumA[blk] / (float)pcntA[blk];
            for (int q = 0; q < pcntA[blk]; ++q) v1[pos++] = mv;
        }
    } else if (t == 32) {
        int top = -1;
        for (int i = 0; i < NN; ++i) {
            float zi = -skey[NN - 1 - i] / EPS;           // z sorted descending
            float yi = zi - (float)(NN - i);              // minus rho
            ++top; psumB[top] = yi; pcntB[top] = 1;
            while (top > 0 &&
                   psumB[top] * (float)pcntB[top - 1] >
                   psumB[top - 1] * (float)pcntB[top]) {
                psumB[top - 1] += psumB[top];
                pcntB[top - 1] += pcntB[top];
                --top;
            }
        }
        int pos = 0;
        for (int blk = 0; blk <= top; ++blk) {
            float mv = psumB[blk] / (float)pcntB[blk];
            for (int q = 0; q < pcntB[blk]; ++q) v2[pos++] = mv;
        }
    }
    __syncthreads();

    // soft_sort output (stays in sorted order)
    out[b * NN + t] = (float)(NN - t) / EPS - v1[t];

    // soft_rank output (scatter through inverse permutation)
    float zdesc = -skey[NN - 1 - t] / EPS;
    int   orig  = sidx[NN - 1 - t];
    out[BB * NN + b * NN + orig] = zdesc - v2[t];
}

// ---------------------------------------------------------------------------
extern "C" void kernel_launch(void* const* d_in, const int* in_sizes, int n_in,
                              void* d_out, int out_size, void* d_ws, size_t ws_size,
                              hipStream_t stream)
{
    const float* cross = (const float*)d_in[0];   // [1, 256]
    const float* init  = (const float*)d_in[1];   // [512, 256]
    const float* W1    = (const float*)d_in[2];   // [1, 32]
    const float* b1    = (const float*)d_in[3];   // [32]
    const float* W2    = (const float*)d_in[4];   // [32, 1]
    const float* b2    = (const float*)d_in[5];   // [1]
    float* out     = (float*)d_out;               // [2 * 512 * 256]
    float* refined = (float*)d_ws;                // N floats scratch

    mlp_refine_wmma<<<1, 512, 0, stream>>>(cross, W1, b1, W2, b2, refined);
    sort_pava_kernel<<<BB, 256, 0, stream>>>(refined, init, out);
}